// YoloLoss_21526376087656
// MI455X (gfx1250) — compile-verified
//
#include <hip/hip_runtime.h>
#include <hip/hip_bf16.h>

// ---------------------------------------------------------------------------
// YOLO loss for MI455X (gfx1250, wave32).
// Kernel 1: fused masked-softplus reduction over the 3 objectness planes of
//           each pyramid level (only ~2 MB of the tensors is ever touched).
//           Per-wave reduction is done on the matrix pipe via
//           V_WMMA_F32_16X16X4_F32 with a ones B-matrix (free compute in a
//           memory/latency-bound kernel; emits the CDNA5 WMMA path).
// Kernel 2: deterministic finalize — per-scale sum/count combine + the tiny
//           positive-box losses (3 boxes x 85 channels spread over threads).
// ---------------------------------------------------------------------------

typedef __attribute__((ext_vector_type(2))) float v2f;
typedef __attribute__((ext_vector_type(8))) float v8f;

__constant__ float c_ANCHORS[3][3][2] = {
    {{10.f, 13.f}, {16.f, 30.f}, {33.f, 23.f}},
    {{30.f, 61.f}, {62.f, 45.f}, {59.f, 119.f}},
    {{116.f, 90.f}, {156.f, 198.f}, {373.f, 326.f}}};

__device__ __forceinline__ float softplusf(float y) {
    // stable: max(y,0) + log1p(exp(-|y|))
    return fmaxf(y, 0.0f) + log1pf(expf(-fabsf(y)));
}

// Sum 32 lane values using one V_WMMA_F32_16X16X4_F32.
// A layout (32-bit 16x4): lanes 0-15 VGPR0->K=0, VGPR1->K=1; lanes 16-31
// VGPR0->K=2, VGPR1->K=3.  With A = {v, 0} and B = ones,
// D[m][n] = p_m + p_{m+16}.  D VGPR r holds row r (lanes 0-15) / row r+8
// (lanes 16-31), so sum of the 8 D regs + shfl_xor(16) = total.
// Requires EXEC all-ones (call only from converged code).
__device__ __forceinline__ float wave_sum_wmma(float v) {
    v2f a; a.x = v;   a.y = 0.0f;
    v2f b; b.x = 1.0f; b.y = 1.0f;
    v8f c = {};
    c = __builtin_amdgcn_wmma_f32_16x16x4_f32(
        /*neg_a=*/false, a, /*neg_b=*/false, b,
        /*c_mod=*/(short)0, c, /*reuse_a=*/false, /*reuse_b=*/false);
    float s = c[0] + c[1] + c[2] + c[3] + c[4] + c[5] + c[6] + c[7];
    s += __shfl_xor(s, 16, 32);
    return s;  // every lane holds the full 32-lane sum
}

__device__ __forceinline__ int wave_sum_i32(int v) {
#pragma unroll
    for (int off = 16; off > 0; off >>= 1) v += __shfl_xor(v, off, 32);
    return v;
}

// Block layout: blocks [0,4) -> scale 0, [4,20) -> scale 1, [20,84) -> scale 2.
__global__ void __launch_bounds__(256)
yolo_neg_reduce(const float* __restrict__ x0, const float* __restrict__ x1,
                const float* __restrict__ x2,
                const int* __restrict__ n0, const int* __restrict__ n1,
                const int* __restrict__ n2,
                float* __restrict__ psum, int* __restrict__ pcnt) {
    const float* x; const int* neg; int gs, base, bid, nblk;
    const int b = blockIdx.x;
    if (b < 4)       { x = x0; neg = n0; gs = 64;  base = 0;   bid = b;      nblk = 4;  }
    else if (b < 20) { x = x1; neg = n1; gs = 128; base = 64;  bid = b - 4;  nblk = 16; }
    else             { x = x2; neg = n2; gs = 256; base = 128; bid = b - 20; nblk = 64; }

    const int plane = gs * gs;
    // objectness channels a*80+4 : a = 0,1,2 -> channels 4, 84, 164 (contiguous planes)
    const float4* p0 = (const float4*)(x + 4   * plane);
    const float4* p1 = (const float4*)(x + 84  * plane);
    const float4* p2 = (const float4*)(x + 164 * plane);
    const int4*   nm = (const int4*)neg;  // neg[3*cell + a] -> 12 ints per 4 cells
    const int groups = plane >> 2;        // float4 groups of cells

    float lsum = 0.0f;
    int   lcnt = 0;
    for (int g = bid * 256 + (int)threadIdx.x; g < groups; g += nblk * 256) {
        float4 v0 = p0[g], v1 = p1[g], v2 = p2[g];
        int4 m0 = nm[3 * g + 0], m1 = nm[3 * g + 1], m2 = nm[3 * g + 2];
        // pair (cell k, anchor a): value {v_a}[k], mask element 3k+a
        float vv[12] = {v0.x, v1.x, v2.x,  v0.y, v1.y, v2.y,
                        v0.z, v1.z, v2.z,  v0.w, v1.w, v2.w};
        int   mm[12] = {m0.x, m0.y, m0.z, m0.w,  m1.x, m1.y, m1.z, m1.w,
                        m2.x, m2.y, m2.z, m2.w};
#pragma unroll
        for (int k = 0; k < 12; ++k) {
            const bool sel = (mm[k] == 0);
            lsum += sel ? softplusf(vv[k]) : 0.0f;
            lcnt += sel ? 1 : 0;
        }
    }

    // converged here: EXEC all-ones -> WMMA wave reduction is legal
    const float wsum = wave_sum_wmma(lsum);
    const int   wcnt = wave_sum_i32(lcnt);

    __shared__ float ssum[8];
    __shared__ int   scnt[8];
    const int lane = threadIdx.x & 31;
    const int wid  = threadIdx.x >> 5;
    if (lane == 0) { ssum[wid] = wsum; scnt[wid] = wcnt; }
    __syncthreads();
    if (threadIdx.x == 0) {
        float bs = 0.0f; int bc = 0;
#pragma unroll
        for (int w = 0; w < 8; ++w) { bs += ssum[w]; bc += scnt[w]; }
        psum[base + bid] = bs;
        pcnt[base + bid] = bc;
    }
}

__global__ void __launch_bounds__(256)
yolo_finalize(const float* __restrict__ psum, const int* __restrict__ pcnt,
              int nb0, int nb1, int nb2,
              const float* __restrict__ x0, const float* __restrict__ x1,
              const float* __restrict__ x2,
              const int* __restrict__ label_cls,
              const float* __restrict__ label_bbx,
              const int* __restrict__ pos_anc, const int* __restrict__ pos_x,
              const int* __restrict__ pos_y,
              const int* __restrict__ scale0, const int* __restrict__ scale1,
              const int* __restrict__ scale2,
              float* __restrict__ out) {
    __shared__ float sloss[3];     // per-scale negative loss (sum/count)
    __shared__ float contrib[256]; // positive-box per-term contributions

    const int lane = threadIdx.x & 31;
    const int wid  = threadIdx.x >> 5;

    // ---- Part A: per-scale combine of block partials (warps 0..2) ----
    if (wid < 3) {
        const int nb = (wid == 0) ? nb0 : (wid == 1) ? nb1 : nb2;
        const float* ps = psum + wid * 64;
        const int*   pc = pcnt + wid * 64;
        float s = 0.0f; int c = 0;
        for (int i = lane; i < nb; i += 32) { s += ps[i]; c += pc[i]; }
#pragma unroll
        for (int off = 16; off > 0; off >>= 1) {
            s += __shfl_xor(s, off, 32);
            c += __shfl_xor(c, off, 32);
        }
        if (lane == 0) sloss[wid] = s / (float)c;
    }

    // ---- Part B: positive boxes, 3 boxes x 85 channels = 255 terms ----
    const int t = threadIdx.x;
    float r = 0.0f;
    if (t < 255) {
        const int j = t / 85;   // box index
        const int c = t % 85;   // channel offset within the anchor's 85-block
        const int ii = (j == 0) ? scale0[0] : (j == 1) ? scale1[0] : scale2[0];
        const float* xi = (ii == 0) ? x0 : (ii == 1) ? x1 : x2;
        const int gsz   = (ii == 0) ? 64 : (ii == 1) ? 128 : 256;
        const int plane = gsz * gsz;
        const int off   = pos_y[j] * gsz + pos_x[j];
        const int a80   = pos_anc[j] * 80;
        const float val = xi[(a80 + c) * plane + off];

        if (c == 4) {
            r = softplusf(-val);                       // BCE(y_obj, 1)
        } else if (c >= 5) {
            r = softplusf(val);                        // BCE(y_cls, one-hot)
            if ((c - 5) == label_cls[j]) r -= val;
            r *= (1.0f / 80.0f);
        } else {
            // smooth-L1 on bbox component c; reference quirks preserved:
            // stride from GRID_SIZES[j] (box index!), anchors[j][ii]
            const float* bb = label_bbx + 4 * j;
            const int   gj     = (j == 0) ? 64 : (j == 1) ? 128 : 256;
            const float stride = 2048.0f / (float)gj;
            const float a0 = c_ANCHORS[j][ii][0];
            const float a1 = c_ANCHORS[j][ii][1];
            float tt;
            if (c == 0)      tt = ((bb[0] + bb[2]) * 0.5f - (float)pos_x[j] * stride) / a0;
            else if (c == 1) tt = ((bb[1] + bb[3]) * 0.5f - (float)pos_y[j] * stride) / a1;
            else if (c == 2) tt = logf((bb[2] - bb[0]) / a0);
            else             tt = logf((bb[3] - bb[1]) / a1);
            const float d = fabsf(val - tt);
            r = ((d < 1.0f) ? 0.5f * d * d : d - 0.5f) * 0.25f;
        }
    }
    contrib[t] = r;
    __syncthreads();

    if (t == 0) {
        float total = sloss[0] + sloss[1] + sloss[2];
        for (int k = 0; k < 255; ++k) total += contrib[k];  // deterministic order
        out[0] = total;
    }
}

extern "C" void kernel_launch(void* const* d_in, const int* in_sizes, int n_in,
                              void* d_out, int out_size, void* d_ws, size_t ws_size,
                              hipStream_t stream) {
    (void)in_sizes; (void)n_in; (void)out_size; (void)ws_size;
    const float* x0        = (const float*)d_in[0];
    const float* x1        = (const float*)d_in[1];
    const float* x2        = (const float*)d_in[2];
    const int*   label_cls = (const int*)d_in[3];
    const float* label_bbx = (const float*)d_in[4];
    const int*   pos_anc   = (const int*)d_in[5];
    const int*   pos_x     = (const int*)d_in[6];
    const int*   pos_y     = (const int*)d_in[7];
    const int*   neg0      = (const int*)d_in[8];
    const int*   neg1      = (const int*)d_in[9];
    const int*   neg2      = (const int*)d_in[10];
    const int*   scale0    = (const int*)d_in[11];
    const int*   scale1    = (const int*)d_in[12];
    const int*   scale2    = (const int*)d_in[13];
    float*       out       = (float*)d_out;

    // Workspace: 3 scale regions x 64 block partials (float sums, int counts).
    float* psum = (float*)d_ws;                          // 192 floats
    int*   pcnt = (int*)((char*)d_ws + 192 * sizeof(float));
    const int nb0 = 4, nb1 = 16, nb2 = 64;               // 84 blocks total

    yolo_neg_reduce<<<nb0 + nb1 + nb2, 256, 0, stream>>>(
        x0, x1, x2, neg0, neg1, neg2, psum, pcnt);
    yolo_finalize<<<1, 256, 0, stream>>>(
        psum, pcnt, nb0, nb1, nb2, x0, x1, x2,
        label_cls, label_bbx, pos_anc, pos_x, pos_y,
        scale0, scale1, scale2, out);
}